// PNA_67482526155042
// MI455X (gfx1250) — compile-verified
//
#include <hip/hip_runtime.h>
#include <math.h>

#define NN 10000
#define EE 640000
#define FF 128
#define EDD 16
#define LL 2
#define KCAT (13 * FF)  // 1664

typedef __attribute__((ext_vector_type(2))) float v2f;
typedef __attribute__((ext_vector_type(8))) float v8f;

// ---------------------------------------------------------------------------
// WMMA helper: D = A(16x4,f32) * B(4x16,f32) + C(16x16,f32), wave32.
// Per-lane fragment layout (CDNA5 ISA 7.12.2):
//   A: row m = lane&15, a.{x,y} = K = k0 + 2*(lane>>4) + {0,1}
//   B: col n = lane&15, b.{x,y} = same K mapping
//   C/D: col n = lane&15, vgpr r -> row r + 8*(lane>>4)
// ---------------------------------------------------------------------------
__device__ __forceinline__ v8f wmma_f32(v2f a, v2f b, v8f c) {
  return __builtin_amdgcn_wmma_f32_16x16x4_f32(false, a, false, b, (short)0, c,
                                               false, false);
}

// ---------------------------------------------------------------------------
// Generic row-major GEMM C(M,128) = A(M,K) @ B(K,128). K % 4 == 0, M % 16 == 0.
// Block = 64 threads (2 waves): one 16-row M-tile, wave w covers cols
// [w*64, w*64+64) as 4 WMMA n-tiles.
// ---------------------------------------------------------------------------
__global__ void __launch_bounds__(64) k_gemm128(const float* __restrict__ A,
                                                const float* __restrict__ B,
                                                float* __restrict__ C, int K) {
  const int lane = threadIdx.x & 31;
  const int wave = threadIdx.x >> 5;
  const int ncol = lane & 15;
  const int koff = (lane >> 4) << 1;
  const int mrow = (blockIdx.x << 4) + (lane & 15);
  const float* __restrict__ Arow = A + (size_t)mrow * (size_t)K;

  v8f acc[4];
#pragma unroll
  for (int t = 0; t < 4; ++t) acc[t] = (v8f)0.0f;

  for (int k0 = 0; k0 < K; k0 += 4) {
    v2f a;
    a.x = Arow[k0 + koff];
    a.y = Arow[k0 + koff + 1];
#pragma unroll
    for (int t = 0; t < 4; ++t) {
      const int nb = ((wave << 2) + t) * 16 + ncol;
      v2f b;
      b.x = B[(size_t)(k0 + koff) * FF + nb];
      b.y = B[(size_t)(k0 + koff + 1) * FF + nb];
      acc[t] = wmma_f32(a, b, acc[t]);
    }
  }

  const int rbase = (blockIdx.x << 4) + ((lane >> 4) << 3);
#pragma unroll
  for (int t = 0; t < 4; ++t) {
    const int col = ((wave << 2) + t) * 16 + ncol;
#pragma unroll
    for (int r = 0; r < 8; ++r)
      C[(size_t)(rbase + r) * FF + col] = acc[t][r];
  }
}

// ---------------------------------------------------------------------------
// Dual GEMM sharing one A pass: C0 = A@B0, C1 = A@B1  (K=128, N=128).
// Used for HS = h@w_src, HD = h@w_dst: halves A traffic, 8 WMMAs/fragment.
// ---------------------------------------------------------------------------
__global__ void __launch_bounds__(64) k_gemm_dual(const float* __restrict__ A,
                                                  const float* __restrict__ B0,
                                                  const float* __restrict__ B1,
                                                  float* __restrict__ C0,
                                                  float* __restrict__ C1) {
  const int lane = threadIdx.x & 31;
  const int wave = threadIdx.x >> 5;
  const int ncol = lane & 15;
  const int koff = (lane >> 4) << 1;
  const int mrow = (blockIdx.x << 4) + (lane & 15);
  const float* __restrict__ Arow = A + (size_t)mrow * FF;

  v8f accS[4], accD[4];
#pragma unroll
  for (int t = 0; t < 4; ++t) { accS[t] = (v8f)0.0f; accD[t] = (v8f)0.0f; }

  for (int k0 = 0; k0 < FF; k0 += 4) {
    v2f a;
    a.x = Arow[k0 + koff];
    a.y = Arow[k0 + koff + 1];
    const size_t r0 = (size_t)(k0 + koff) * FF;
    const size_t r1 = (size_t)(k0 + koff + 1) * FF;
#pragma unroll
    for (int t = 0; t < 4; ++t) {
      const int nb = ((wave << 2) + t) * 16 + ncol;
      v2f b0; b0.x = B0[r0 + nb]; b0.y = B0[r1 + nb];
      accS[t] = wmma_f32(a, b0, accS[t]);
      v2f b1; b1.x = B1[r0 + nb]; b1.y = B1[r1 + nb];
      accD[t] = wmma_f32(a, b1, accD[t]);
    }
  }

  const int rbase = (blockIdx.x << 4) + ((lane >> 4) << 3);
#pragma unroll
  for (int t = 0; t < 4; ++t) {
    const int col = ((wave << 2) + t) * 16 + ncol;
#pragma unroll
    for (int r = 0; r < 8; ++r) {
      C0[(size_t)(rbase + r) * FF + col] = accS[t][r];
      C1[(size_t)(rbase + r) * FF + col] = accD[t][r];
    }
  }
}

// ---------------------------------------------------------------------------
// Small vector-matrix: out[n] = sum_k v[k] * W[k,n] + b[n]  (128x128 W)
// ---------------------------------------------------------------------------
__global__ void k_vecmm(const float* __restrict__ v, const float* __restrict__ W,
                        const float* __restrict__ b, float* __restrict__ outv) {
  const int n = threadIdx.x;
  float acc = b[n];
#pragma unroll 8
  for (int k = 0; k < FF; ++k) acc += v[k] * W[k * FF + n];
  outv[n] = acc;
}

// ---------------------------------------------------------------------------
// Degree / scalar stats
// ---------------------------------------------------------------------------
__global__ void k_deg(const long long* __restrict__ ei, float* __restrict__ deg,
                      int E) {
  const int e = blockIdx.x * blockDim.x + threadIdx.x;
  if (e >= E) return;
  const int d = (int)ei[2 * e + 1];
  atomicAdd(&deg[d], 1.0f);
}

__global__ void k_avglog(const long long* __restrict__ dh,
                         float* __restrict__ out) {
  __shared__ float s1[64], s2[64];
  const int t = threadIdx.x;
  const float d = (float)dh[t];
  s1[t] = logf((float)t + 1.0f) * d;
  s2[t] = d;
  __syncthreads();
  for (int o = 32; o > 0; o >>= 1) {
    if (t < o) { s1[t] += s1[t + o]; s2[t] += s2[t + o]; }
    __syncthreads();
  }
  if (t == 0) *out = s1[0] / s2[0];
}

__global__ void k_ampatt(const float* __restrict__ deg,
                         const float* __restrict__ avg, float* __restrict__ amp,
                         float* __restrict__ att, int n) {
  const int i = blockIdx.x * blockDim.x + threadIdx.x;
  if (i >= n) return;
  const float d = deg[i];
  const float ld = logf(d + 1.0f);
  const float av = *avg;
  amp[i] = ld / av;
  att[i] = (d == 0.0f) ? 1.0f : av / ld;
}

// ---------------------------------------------------------------------------
// Order-preserving float <-> uint map so segment min/max use native U32 atomics
// ---------------------------------------------------------------------------
__device__ __forceinline__ unsigned mapf(float x) {
  const unsigned u = __float_as_uint(x);
  return (u & 0x80000000u) ? ~u : (u | 0x80000000u);
}
__device__ __forceinline__ float unmapf(unsigned v) {
  return __uint_as_float((v & 0x80000000u) ? (v ^ 0x80000000u) : ~v);
}

// ---------------------------------------------------------------------------
// Edge phase: one wave per edge, lane owns 4 features. The edge id is forced
// scalar via readfirstlane so edge_index / edge_weight / 16 edge features
// become SMEM s_loads (one broadcast instead of 32 VMEM requests) and the
// gather/scatter addresses become scalar-base + lane-offset.
// ---------------------------------------------------------------------------
__global__ void __launch_bounds__(256) k_edge(
    const long long* __restrict__ ei, const float* __restrict__ ef,
    const float* __restrict__ ew, const float* __restrict__ HS,
    const float* __restrict__ HD, const float* __restrict__ Wc,
    const float* __restrict__ bc, float* __restrict__ SUM,
    float* __restrict__ SUMSQ, unsigned* __restrict__ MN,
    unsigned* __restrict__ MX, int E) {
  const int gid = blockIdx.x * blockDim.x + threadIdx.x;
  const int e = __builtin_amdgcn_readfirstlane(gid >> 5);  // wave-uniform
  if (e >= E) return;
  const int c = (threadIdx.x & 31) << 2;  // 4 consecutive features per lane

  const long long s = ei[2 * (size_t)e];      // scalar loads
  const long long d = ei[2 * (size_t)e + 1];
  const float w = ew[e];
  const float* __restrict__ efp = ef + (size_t)e * EDD;

  const float4 b4 = *(const float4*)(bc + c);
  float ax = b4.x, ay = b4.y, az = b4.z, aw = b4.w;
#pragma unroll
  for (int k = 0; k < EDD; ++k) {
    const float v = efp[k];  // scalar (uniform) load
    const float4 wr = *(const float4*)(Wc + k * FF + c);
    ax += v * wr.x; ay += v * wr.y; az += v * wr.z; aw += v * wr.w;
  }
  const float4 hs = *(const float4*)(HS + (size_t)s * FF + c);
  const float4 hd = *(const float4*)(HD + (size_t)d * FF + c);
  const float m0 = (ax + hs.x + hd.x) * w;
  const float m1 = (ay + hs.y + hd.y) * w;
  const float m2 = (az + hs.z + hd.z) * w;
  const float m3 = (aw + hs.w + hd.w) * w;

  const size_t base = (size_t)d * FF + c;
  atomicAdd(&SUM[base + 0], m0);
  atomicAdd(&SUM[base + 1], m1);
  atomicAdd(&SUM[base + 2], m2);
  atomicAdd(&SUM[base + 3], m3);
  atomicAdd(&SUMSQ[base + 0], m0 * m0);
  atomicAdd(&SUMSQ[base + 1], m1 * m1);
  atomicAdd(&SUMSQ[base + 2], m2 * m2);
  atomicAdd(&SUMSQ[base + 3], m3 * m3);
  atomicMin(&MN[base + 0], mapf(m0));
  atomicMin(&MN[base + 1], mapf(m1));
  atomicMin(&MN[base + 2], mapf(m2));
  atomicMin(&MN[base + 3], mapf(m3));
  atomicMax(&MX[base + 0], mapf(m0));
  atomicMax(&MX[base + 1], mapf(m1));
  atomicMax(&MX[base + 2], mapf(m2));
  atomicMax(&MX[base + 3], mapf(m3));
}

// ---------------------------------------------------------------------------
// In-place finalize: SUM->mean, SUMSQ->std, MN/MX -> floats (0 if no nbrs)
// ---------------------------------------------------------------------------
__global__ void k_finalize(const float* __restrict__ deg, float* __restrict__ SUM,
                           float* __restrict__ SUMSQ, unsigned* __restrict__ MN,
                           unsigned* __restrict__ MX, int total) {
  const int i = blockIdx.x * blockDim.x + threadIdx.x;
  if (i >= total) return;
  const int node = i >> 7;
  const float d = deg[node];
  const float sd = fmaxf(d, 1.0f);
  const float mean = SUM[i] / sd;
  const float ms = SUMSQ[i] / sd;
  const float var = ms - mean * mean;
  const float stdv = sqrtf(fmaxf(var, 0.0f) + 1e-5f);
  const bool hn = d > 0.0f;
  const float mnv = hn ? unmapf(MN[i]) : 0.0f;
  const float mxv = hn ? unmapf(MX[i]) : 0.0f;
  SUM[i] = mean;
  SUMSQ[i] = stdv;
  ((float*)MN)[i] = mnv;
  ((float*)MX)[i] = mxv;
}

// ---------------------------------------------------------------------------
// Fused post+update GEMM. Reference computes
//   out = concat([h, agg, agg*amp, agg*att]) @ post_w + post_b, then @ upd_w.
// With PWU = post_w @ upd_w pre-folded, and amp/att being per-ROW scalars:
//   out = h@PWU[0:128] + agg@PWU[128:640]
//       + amp ⊙ (agg@PWU[640:1152]) + att ⊙ (agg@PWU[1152:1664]) + btot
// Three accumulator sets share each agg A-fragment (12 WMMAs per K-step);
// K loop runs 640 instead of 1664 -> 2.6x fewer A-fragment loads. Row scaling
// by amp/att is applied in the epilogue.
// agg k-blocks: 0:mean 1:mn 2:mx 3:std (each 128 wide).
// ---------------------------------------------------------------------------
__global__ void __launch_bounds__(64) k_postupd(
    const float* __restrict__ h, const float* __restrict__ mean_,
    const float* __restrict__ mn_, const float* __restrict__ mx_,
    const float* __restrict__ std_, const float* __restrict__ amp,
    const float* __restrict__ att, const float* __restrict__ PWU,
    const float* __restrict__ btot, float* __restrict__ out) {
  const int lane = threadIdx.x & 31;
  const int wave = threadIdx.x >> 5;
  const int ncol = lane & 15;
  const int koff = (lane >> 4) << 1;
  const int m = (blockIdx.x << 4) + (lane & 15);
  const int rowoff = m * FF;

  v8f accB[4], accA[4], accT[4];
#pragma unroll
  for (int t = 0; t < 4; ++t) {
    accB[t] = (v8f)0.0f; accA[t] = (v8f)0.0f; accT[t] = (v8f)0.0f;
  }

  // Phase 1: h block (K rows 0..127 of PWU) -> base accumulator only.
  for (int k0 = 0; k0 < FF; k0 += 4) {
    v2f a;
    a.x = h[rowoff + k0 + koff];
    a.y = h[rowoff + k0 + koff + 1];
#pragma unroll
    for (int t = 0; t < 4; ++t) {
      const int nb = ((wave << 2) + t) * 16 + ncol;
      v2f b;
      b.x = PWU[(size_t)(k0 + koff) * FF + nb];
      b.y = PWU[(size_t)(k0 + koff + 1) * FF + nb];
      accB[t] = wmma_f32(a, b, accB[t]);
    }
  }

  // Phase 2: agg blocks; each A-fragment feeds base/amp/att accumulators.
  const float* __restrict__ G[4] = {mean_, mn_, mx_, std_};
#pragma unroll
  for (int g = 0; g < 4; ++g) {
    const float* __restrict__ Gp = G[g] + rowoff;
    for (int k0 = 0; k0 < FF; k0 += 4) {
      v2f a;
      a.x = Gp[k0 + koff];
      a.y = Gp[k0 + koff + 1];
      const int kk0 = g * FF + k0 + koff;      // index within agg (0..511)
      const size_t rB0 = (size_t)(FF + kk0) * FF;
      const size_t rB1 = (size_t)(FF + kk0 + 1) * FF;
      const size_t rA0 = (size_t)(5 * FF + kk0) * FF;
      const size_t rA1 = (size_t)(5 * FF + kk0 + 1) * FF;
      const size_t rT0 = (size_t)(9 * FF + kk0) * FF;
      const size_t rT1 = (size_t)(9 * FF + kk0 + 1) * FF;
#pragma unroll
      for (int t = 0; t < 4; ++t) {
        const int nb = ((wave << 2) + t) * 16 + ncol;
        v2f b;
        b.x = PWU[rB0 + nb]; b.y = PWU[rB1 + nb];
        accB[t] = wmma_f32(a, b, accB[t]);
        b.x = PWU[rA0 + nb]; b.y = PWU[rA1 + nb];
        accA[t] = wmma_f32(a, b, accA[t]);
        b.x = PWU[rT0 + nb]; b.y = PWU[rT1 + nb];
        accT[t] = wmma_f32(a, b, accT[t]);
      }
    }
  }

  // Epilogue: out[row,col] = accB + amp[row]*accA + att[row]*accT + btot[col]
  const int rbase = (blockIdx.x << 4) + ((lane >> 4) << 3);
  float ampr[8], attr[8];
#pragma unroll
  for (int r = 0; r < 8; ++r) {
    ampr[r] = amp[rbase + r];
    attr[r] = att[rbase + r];
  }
#pragma unroll
  for (int t = 0; t < 4; ++t) {
    const int col = ((wave << 2) + t) * 16 + ncol;
    const float bv = btot[col];
#pragma unroll
    for (int r = 0; r < 8; ++r)
      out[(size_t)(rbase + r) * FF + col] =
          accB[t][r] + ampr[r] * accA[t][r] + attr[r] * accT[t][r] + bv;
  }
}

// ---------------------------------------------------------------------------
// Host orchestration
// ---------------------------------------------------------------------------
extern "C" void kernel_launch(void* const* d_in, const int* in_sizes, int n_in,
                              void* d_out, int out_size, void* d_ws,
                              size_t ws_size, hipStream_t stream) {
  const float* x = (const float*)d_in[0];
  const float* ef = (const float*)d_in[1];
  const float* ew = (const float*)d_in[2];
  const float* eew = (const float*)d_in[3];   // (L,16,128)
  const float* eeb = (const float*)d_in[4];   // (L,128)
  const float* prw = (const float*)d_in[5];   // (L,384,128)
  const float* prb = (const float*)d_in[6];   // (L,128)
  const float* pow_ = (const float*)d_in[7];  // (L,1664,128)
  const float* pob = (const float*)d_in[8];   // (L,128)
  const float* upw = (const float*)d_in[9];   // (L,128,128)
  const float* upb = (const float*)d_in[10];  // (L,128)
  const long long* eidx = (const long long*)d_in[11];  // (E,2) int64
  const long long* dh = (const long long*)d_in[12];    // (64,) int64
  float* out = (float*)d_out;

  char* w = (char*)d_ws;
  size_t off = 0;
  auto alloc = [&](size_t bytes) -> char* {
    char* p = w + off;
    off = (off + bytes + 255) & ~(size_t)255;
    return p;
  };
  float* deg = (float*)alloc((size_t)NN * 4);
  float* amp = (float*)alloc((size_t)NN * 4);
  float* att = (float*)alloc((size_t)NN * 4);
  float* avg = (float*)alloc(4);
  float* HS = (float*)alloc((size_t)NN * FF * 4);
  float* HD = (float*)alloc((size_t)NN * FF * 4);
  float* SUM = (float*)alloc((size_t)NN * FF * 4);
  float* SUMSQ = (float*)alloc((size_t)NN * FF * 4);
  unsigned* MN = (unsigned*)alloc((size_t)NN * FF * 4);
  unsigned* MX = (unsigned*)alloc((size_t)NN * FF * 4);
  float* Wc = (float*)alloc((size_t)EDD * FF * 4);
  float* bc = (float*)alloc((size_t)FF * 4);
  float* PWU = (float*)alloc((size_t)KCAT * FF * 4);
  float* btot = (float*)alloc((size_t)FF * 4);
  float* Hbuf = (float*)alloc((size_t)NN * FF * 4);

  // ---- Degree statistics (independent of layer) ----
  hipMemsetAsync(deg, 0, (size_t)NN * 4, stream);
  k_deg<<<(EE + 255) / 256, 256, 0, stream>>>(eidx, deg, EE);
  k_avglog<<<1, 64, 0, stream>>>(dh, avg);
  k_ampatt<<<(NN + 255) / 256, 256, 0, stream>>>(deg, avg, amp, att, NN);

  for (int l = 0; l < LL; ++l) {
    const float* h_in = (l == 0) ? x : Hbuf;
    float* h_out = (l == LL - 1) ? out : Hbuf;
    const float* w_src = prw + (size_t)l * 3 * FF * FF;
    const float* w_dst = w_src + (size_t)FF * FF;
    const float* w_ea = w_src + (size_t)2 * FF * FF;

    // Fold edge-encoder into the message weights: Wc = W_enc @ w_ea (16x128)
    k_gemm128<<<EDD / 16, 64, 0, stream>>>(eew + (size_t)l * EDD * FF, w_ea, Wc,
                                           FF);
    // bc = b_enc @ w_ea + pre_b
    k_vecmm<<<1, FF, 0, stream>>>(eeb + (size_t)l * FF, w_ea,
                                  prb + (size_t)l * FF, bc);

    // Node projections: HS = h @ w_src, HD = h @ w_dst in one A pass
    k_gemm_dual<<<NN / 16, 64, 0, stream>>>(h_in, w_src, w_dst, HS, HD);

    // Edge aggregation phase
    hipMemsetAsync(SUM, 0, (size_t)NN * FF * 4, stream);
    hipMemsetAsync(SUMSQ, 0, (size_t)NN * FF * 4, stream);
    hipMemsetAsync(MN, 0xFF, (size_t)NN * FF * 4, stream);  // +max uint
    hipMemsetAsync(MX, 0x00, (size_t)NN * FF * 4, stream);  // min uint
    k_edge<<<(EE * 32 + 255) / 256, 256, 0, stream>>>(eidx, ef, ew, HS, HD, Wc,
                                                      bc, SUM, SUMSQ, MN, MX,
                                                      EE);
    k_finalize<<<(NN * FF + 255) / 256, 256, 0, stream>>>(deg, SUM, SUMSQ, MN,
                                                          MX, NN * FF);

    // Fold update GEMM into post GEMM: PWU = post_w @ upd_w (1664x128)
    k_gemm128<<<KCAT / 16, 64, 0, stream>>>(pow_ + (size_t)l * KCAT * FF,
                                            upw + (size_t)l * FF * FF, PWU, FF);
    // btot = post_b @ upd_w + upd_b
    k_vecmm<<<1, FF, 0, stream>>>(pob + (size_t)l * FF,
                                  upw + (size_t)l * FF * FF,
                                  upb + (size_t)l * FF, btot);

    // Fused post+update GEMM (3 accumulator sets, amp/att in epilogue)
    k_postupd<<<NN / 16, 64, 0, stream>>>(h_in, SUM, (const float*)MN,
                                          (const float*)MX, SUMSQ, amp, att,
                                          PWU, btot, h_out);
  }
}